// DiffusionLoss_90649579749650
// MI455X (gfx1250) — compile-verified
//
#include <hip/hip_runtime.h>
#include <hip/hip_bf16.h>

// Fused L1 + SSIM loss for (8192,3,32,32) fp32 NCHW pairs.
// Separable 11-tap Gaussian blur done as two banded 32x32x32 fp32 matmuls per
// (image,channel) tile on the CDNA5 matrix pipe (V_WMMA_F32_16X16X4_F32),
// keeping full fp32 precision (sigma = blur(x^2) - mu^2 cancels badly in low
// precision).
// Optimizations:
//  * sum/difference transform: with u=x+y, v=x-y only FOUR blurs are needed
//    (U=blur(u), V=blur(v), P=blur(u^2), Q=blur(v^2)):
//      mu1*mu2=(U^2-V^2)/4, mu1^2+mu2^2=(U^2+V^2)/2,
//      Sxy=(P-Q)/4,         Sxx+Syy=(P+Q)/2.
//  * banded Gaussian: all-zero 16x4 K-blocks skipped (44 wmma per field
//    instead of 64) with bit-identical results.
//  * zero-padded invalid rows/cols -> outputs outside the 22x22 VALID region
//    are exactly zero and are masked in the reduction.

typedef float v2f __attribute__((ext_vector_type(2)));
typedef float v8f __attribute__((ext_vector_type(8)));

#define WAVES_PER_WG   2
#define SLABS_PER_WAVE 4
#define N_SLABS        (8192 * 3)                                   // 24576
#define NWG            (N_SLABS / (SLABS_PER_WAVE * WAVES_PER_WG))  // 3072
#define LSTRIDE        34                     // LDS row stride (floats): bank-conflict free
#define BUF_F          (32 * LSTRIDE)         // 1088 floats per 32x32 tile buffer
#define NBUF           5                      // buffers 0..3 = U,V,P,Q; 4 = pass-1 scratch

#define L1_COUNT   25165824.0f                // 8192*3*32*32
#define SSIM_COUNT 11894784.0f                // 8192*3*22*22

static __device__ __forceinline__ v8f wmma4(v2f a, v2f b, v8f c) {
  // D(16x16,f32) = A(16x4,f32) * B(4x16,f32) + C
  return __builtin_amdgcn_wmma_f32_16x16x4_f32(false, a, false, b, (short)0, c,
                                               false, false);
}

static __device__ __forceinline__ float gauss_raw(int d) {
  float t = (float)(d - 5);
  return expf(-t * t / 4.5f);  // 2*sigma^2 = 4.5
}

__global__ __launch_bounds__(WAVES_PER_WG * 32)
void fused_l1_ssim_stage1(const float* __restrict__ X,
                          const float* __restrict__ Y,
                          float* __restrict__ partial) {
  __shared__ float lds[WAVES_PER_WG * NBUF * BUF_F];  // ~43.5 KB
  __shared__ float red[WAVES_PER_WG][2];

  const int tid  = threadIdx.x;
  const int wave = tid >> 5;
  const int lane = tid & 31;
  const int lo   = lane & 15;   // N / M index inside 16-wide half
  const int hi   = lane >> 4;   // which half of the wave
  float* myb = lds + wave * (NBUF * BUF_F);

  // ---- Gaussian weights (normalized, fp32, matches reference) ----
  float gsum = 0.f;
#pragma unroll
  for (int d = 0; d < 11; ++d) gsum += gauss_raw(d);
  const float invs = 1.f / gsum;

  // ---- A-fragments of G (22x32 banded, zero rows >=22), pass 1 ----
  // A layout (16x4 fp32): lane(lo) = row m, float2 = K cols {4k+2hi, 4k+2hi+1}
  v2f ga[2][8];
#pragma unroll
  for (int mt = 0; mt < 2; ++mt) {
    int m = 16 * mt + lo;
#pragma unroll
    for (int k = 0; k < 8; ++k) {
      int c0 = 4 * k + 2 * hi;
      int d0 = c0 - m, d1 = c0 + 1 - m;
      float e0 = 0.f, e1 = 0.f;
      if (m < 22 && d0 >= 0 && d0 <= 10) e0 = gauss_raw(d0) * invs;
      if (m < 22 && d1 >= 0 && d1 <= 10) e1 = gauss_raw(d1) * invs;
      v2f v; v.x = e0; v.y = e1;
      ga[mt][k] = v;
    }
  }

  // ---- B-fragments of G^T (32x22, zero cols >=22), pass 2 ----
  // B layout (4x16 fp32): lane(lo) = col c', float2 = K rows {4k+2hi, 4k+2hi+1}
  // G^T[c][c'] = g[c - c']
  v2f gtb[8][2];
#pragma unroll
  for (int k = 0; k < 8; ++k) {
    int c0 = 4 * k + 2 * hi;
#pragma unroll
    for (int nt = 0; nt < 2; ++nt) {
      int cp = 16 * nt + lo;
      int d0 = c0 - cp, d1 = c0 + 1 - cp;
      float e0 = 0.f, e1 = 0.f;
      if (cp < 22 && d0 >= 0 && d0 <= 10) e0 = gauss_raw(d0) * invs;
      if (cp < 22 && d1 >= 0 && d1 <= 10) e1 = gauss_raw(d1) * invs;
      v2f v; v.x = e0; v.y = e1;
      gtb[k][nt] = v;
    }
  }

  const int gwave = blockIdx.x * WAVES_PER_WG + wave;
  float l1_sum   = 0.f;
  float ssim_sum = 0.f;
  const float C1 = 1e-4f;  // (0.01*1.0)^2
  const float C2 = 9e-4f;  // (0.03*1.0)^2

#pragma unroll 1
  for (int s = 0; s < SLABS_PER_WAVE; ++s) {
    const int slab = gwave * SLABS_PER_WAVE + s;
    const float* xp = X + (size_t)slab * 1024;
    const float* yp = Y + (size_t)slab * 1024;

    // ---- load X,Y into B-fragment layout as u=x+y, v=x-y; accumulate L1 ----
    // lane holds cols {lo, lo+16}, rows 4k+2hi and 4k+2hi+1: exact partition.
    v2f uf[8][2], vf[8][2];
#pragma unroll
    for (int k = 0; k < 8; ++k) {
      int r = 4 * k + 2 * hi;
#pragma unroll
      for (int nt = 0; nt < 2; ++nt) {
        int c = 16 * nt + lo;
        float x0 = xp[r * 32 + c], x1 = xp[(r + 1) * 32 + c];
        float y0 = yp[r * 32 + c], y1 = yp[(r + 1) * 32 + c];
        float d0 = x0 - y0, d1 = x1 - y1;
        v2f vu; vu.x = x0 + y0; vu.y = x1 + y1;
        v2f vv; vv.x = d0;      vv.y = d1;
        uf[k][nt] = vu;
        vf[k][nt] = vv;
        l1_sum += fabsf(d0) + fabsf(d1);
      }
    }

    // ---- 4 fields: U=blur(u), V=blur(v), P=blur(u^2), Q=blur(v^2) ----
#pragma unroll
    for (int f = 0; f < 4; ++f) {
      // pass 1: Vt = G * field   (22x32 valid, rest exactly zero)
      // Banded G: rows 0-15 touch K-cols 0-25 (k<=6); rows 16-21 touch
      // K-cols 16-31 (k>=4). Other blocks are exactly zero -> skipped.
      v8f acc[2][2];
#pragma unroll
      for (int mt = 0; mt < 2; ++mt)
#pragma unroll
        for (int nt = 0; nt < 2; ++nt)
          acc[mt][nt] = (v8f){0.f, 0.f, 0.f, 0.f, 0.f, 0.f, 0.f, 0.f};

#pragma unroll
      for (int k = 0; k < 8; ++k) {
        v2f b0, b1;
        v2f u0 = uf[k][0], u1 = uf[k][1];
        v2f v0 = vf[k][0], v1 = vf[k][1];
        if (f == 0)      { b0 = u0;      b1 = u1;      }
        else if (f == 1) { b0 = v0;      b1 = v1;      }
        else if (f == 2) { b0 = u0 * u0; b1 = u1 * u1; }
        else             { b0 = v0 * v0; b1 = v1 * v1; }
        if (k <= 6) {
          acc[0][0] = wmma4(ga[0][k], b0, acc[0][0]);
          acc[0][1] = wmma4(ga[0][k], b1, acc[0][1]);
        }
        if (k >= 4) {
          acc[1][0] = wmma4(ga[1][k], b0, acc[1][0]);
          acc[1][1] = wmma4(ga[1][k], b1, acc[1][1]);
        }
      }

      // D-store pass-1 result to LDS scratch (row-major, stride 34)
      float* Vt = myb + 4 * BUF_F;
#pragma unroll
      for (int mt = 0; mt < 2; ++mt)
#pragma unroll
        for (int nt = 0; nt < 2; ++nt)
#pragma unroll
          for (int j = 0; j < 8; ++j)
            Vt[(16 * mt + j + 8 * hi) * LSTRIDE + 16 * nt + lo] = acc[mt][nt][j];
      __syncthreads();

      // pass 2: H = Vt * G^T
      // Banded G^T: output cols 0-15 need K-rows 0-25 (k<=6); output cols
      // 16-21 need K-rows 16-31 (k>=4). Other blocks exactly zero -> skipped.
      v8f h[2][2];
#pragma unroll
      for (int mt = 0; mt < 2; ++mt)
#pragma unroll
        for (int nt = 0; nt < 2; ++nt)
          h[mt][nt] = (v8f){0.f, 0.f, 0.f, 0.f, 0.f, 0.f, 0.f, 0.f};

#pragma unroll
      for (int k = 0; k < 8; ++k) {
        int col = 4 * k + 2 * hi;
        int i0 = lo * LSTRIDE + col;         // A-frag row = lo (tile mt=0)
        int i1 = (16 + lo) * LSTRIDE + col;  // tile mt=1
        v2f va0; va0.x = Vt[i0]; va0.y = Vt[i0 + 1];
        v2f va1; va1.x = Vt[i1]; va1.y = Vt[i1 + 1];
        if (k <= 6) {
          h[0][0] = wmma4(va0, gtb[k][0], h[0][0]);
          h[1][0] = wmma4(va1, gtb[k][0], h[1][0]);
        }
        if (k >= 4) {
          h[0][1] = wmma4(va0, gtb[k][1], h[0][1]);
          h[1][1] = wmma4(va1, gtb[k][1], h[1][1]);
        }
      }

      // D-store blurred field to its LDS buffer
      float* H = myb + f * BUF_F;
#pragma unroll
      for (int mt = 0; mt < 2; ++mt)
#pragma unroll
        for (int nt = 0; nt < 2; ++nt)
#pragma unroll
          for (int j = 0; j < 8; ++j)
            H[(16 * mt + j + 8 * hi) * LSTRIDE + 16 * nt + lo] = h[mt][nt][j];
      __syncthreads();
    }

    // ---- SSIM map over the valid 22x22 region ----
    for (int p = lane; p < 484; p += 32) {
      int r = p / 22;
      int c = p - 22 * r;
      int idx = r * LSTRIDE + c;
      float Uv = myb[0 * BUF_F + idx];
      float Vv = myb[1 * BUF_F + idx];
      float Pv = myb[2 * BUF_F + idx];
      float Qv = myb[3 * BUF_F + idx];
      float U2 = Uv * Uv, V2 = Vv * Vv;
      float m12 = 0.25f * (U2 - V2);          // mu1*mu2
      float msq = 0.5f  * (U2 + V2);          // mu1^2 + mu2^2
      float s12 = 0.25f * (Pv - Qv) - m12;    // sigma12
      float ssq = 0.5f  * (Pv + Qv) - msq;    // sigma1^2 + sigma2^2
      float num = (2.f * m12 + C1) * (2.f * s12 + C2);
      float den = (msq + C1) * (ssq + C2);
      ssim_sum += num / den;
    }
    __syncthreads();  // field buffers consumed before next slab overwrites them
  }

  // ---- deterministic reduction: wave -> WG partial ----
#pragma unroll
  for (int off = 16; off > 0; off >>= 1) {
    l1_sum   += __shfl_xor(l1_sum, off, 32);
    ssim_sum += __shfl_xor(ssim_sum, off, 32);
  }
  if (lane == 0) { red[wave][0] = l1_sum; red[wave][1] = ssim_sum; }
  __syncthreads();
  if (tid == 0) {
    float a = 0.f, b = 0.f;
#pragma unroll
    for (int w = 0; w < WAVES_PER_WG; ++w) { a += red[w][0]; b += red[w][1]; }
    partial[2 * blockIdx.x]     = a;
    partial[2 * blockIdx.x + 1] = b;
  }
}

__global__ __launch_bounds__(256)
void fused_l1_ssim_stage2(const float* __restrict__ partial,
                          float* __restrict__ out) {
  __shared__ float sl[256], ss[256];
  const int t = threadIdx.x;
  float a = 0.f, b = 0.f;
  for (int i = t; i < NWG; i += 256) {
    a += partial[2 * i];
    b += partial[2 * i + 1];
  }
  sl[t] = a; ss[t] = b;
  __syncthreads();
  for (int off = 128; off > 0; off >>= 1) {
    if (t < off) { sl[t] += sl[t + off]; ss[t] += ss[t + off]; }
    __syncthreads();
  }
  if (t == 0) {
    float l1m   = sl[0] / L1_COUNT;
    float ssimm = ss[0] / SSIM_COUNT;
    out[0] = l1m + 0.1f * (1.f - ssimm);
  }
}

extern "C" void kernel_launch(void* const* d_in, const int* in_sizes, int n_in,
                              void* d_out, int out_size, void* d_ws, size_t ws_size,
                              hipStream_t stream) {
  const float* X = (const float*)d_in[0];  // predicted
  const float* Y = (const float*)d_in[1];  // target
  float* partial = (float*)d_ws;           // NWG*2 floats (written, never read stale)
  fused_l1_ssim_stage1<<<NWG, WAVES_PER_WG * 32, 0, stream>>>(X, Y, partial);
  fused_l1_ssim_stage2<<<1, 256, 0, stream>>>(partial, (float*)d_out);
}